// ScaleGraphFormerAttentionLayer_54417235640722
// MI455X (gfx1250) — compile-verified
//
#include <hip/hip_runtime.h>
#include <math.h>

#define N_NODES 50000
#define N_EDGES 400000
#define IN_DIM  128
#define NH      8
#define HD      16
#define CLAMPV  5.0f

typedef float v2f __attribute__((ext_vector_type(2)));
typedef float v8f __attribute__((ext_vector_type(8)));

// Full-precision fp32 WMMA: D(16x16,f32) = A(16x4,f32) * B(4x16,f32) + C
__device__ __forceinline__ v8f wmma_f32_k4(v2f a, v2f b, v8f c) {
    return __builtin_amdgcn_wmma_f32_16x16x4_f32(
        /*neg_a=*/false, a, /*neg_b=*/false, b,
        /*c_mod=*/(short)0, c, /*reuse_a=*/false, /*reuse_b=*/false);
}

// ---------------------------------------------------------------------------
// Zero scratch regions (ws is poisoned 0xAA by the harness).
// ---------------------------------------------------------------------------
__global__ __launch_bounds__(256)
void zero_kernel(float* __restrict__ p, long long n) {
    long long i = (long long)blockIdx.x * blockDim.x + threadIdx.x;
    if (i < n) p[i] = 0.0f;
}

// ---------------------------------------------------------------------------
// Q/K/V node projections: [Q|K|V](N x 384) = x(N x 128) @ W + b
// 1 block = 8 waves; wave w computes 16-col tiles 3w..3w+2 of 24.
// ---------------------------------------------------------------------------
__global__ __launch_bounds__(256)
void qkv_gemm(const float* __restrict__ x,
              const float* __restrict__ Qw, const float* __restrict__ Qb,
              const float* __restrict__ Kw, const float* __restrict__ Kb,
              const float* __restrict__ Vw, const float* __restrict__ Vb,
              float* __restrict__ Qo, float* __restrict__ Ko, float* __restrict__ Vo)
{
    const int lane = threadIdx.x & 31;
    const int wave = threadIdx.x >> 5;
    const int half = lane >> 4;     // selects K pair within the fragment
    const int l16  = lane & 15;
    const int row0 = blockIdx.x * 16;

    const float* W[3]; const float* Bv[3]; float* Out[3]; int col0[3];
#pragma unroll
    for (int t = 0; t < 3; ++t) {
        int j0 = (wave * 3 + t) * 16;            // global col 0..368
        int m  = j0 >> 7;                        // 0:Q 1:K 2:V
        W[t]   = (m == 0) ? Qw : (m == 1) ? Kw : Vw;
        Bv[t]  = (m == 0) ? Qb : (m == 1) ? Kb : Vb;
        Out[t] = (m == 0) ? Qo : (m == 1) ? Ko : Vo;
        col0[t] = j0 & 127;
    }

    v8f acc[3] = {};
    const int arow = row0 + l16;
    for (int k = 0; k < IN_DIM; k += 4) {
        const int ka = k + half * 2;
        v2f a;
        a.x = x[arow * IN_DIM + ka];
        a.y = x[arow * IN_DIM + ka + 1];
#pragma unroll
        for (int t = 0; t < 3; ++t) {
            v2f b;
            b.x = W[t][(ka    ) * IN_DIM + col0[t] + l16];
            b.y = W[t][(ka + 1) * IN_DIM + col0[t] + l16];
            acc[t] = wmma_f32_k4(a, b, acc[t]);
        }
    }
#pragma unroll
    for (int t = 0; t < 3; ++t) {
        const int c   = col0[t] + l16;
        const float bias = Bv[t][c];
#pragma unroll
        for (int r = 0; r < 8; ++r) {
            const int row = row0 + r + half * 8;
            Out[t][row * IN_DIM + c] = acc[t][r] + bias;
        }
    }
}

// ---------------------------------------------------------------------------
// Fused edge kernel: per 16-edge tile, wave w == head h.
//   Ee slice (16 x 32) for head h via two WMMA accumulators (E_w | E_b),
//   then gather K[src]/Q[dst], signed-sqrt, write wE (into d_out),
//   score = exp(clamp(e_t . Aw)), atomic ssum / deg.
// (segment-max pass dropped: scores clamped to +-5, unshifted softmax is
//  equivalent to ~1e-16.)
// ---------------------------------------------------------------------------
__global__ __launch_bounds__(256)
void edge_kernel(const float* __restrict__ ea,
                 const int*   __restrict__ src,
                 const int*   __restrict__ dst,
                 const float* __restrict__ Ew, const float* __restrict__ Eb,
                 const float* __restrict__ Aw,
                 const float* __restrict__ Q,  const float* __restrict__ K,
                 float* __restrict__ wE,
                 float* __restrict__ score,
                 float* __restrict__ ssum,
                 float* __restrict__ deg)
{
    const int lane = threadIdx.x & 31;
    const int h    = threadIdx.x >> 5;          // wave index == head
    const int half = lane >> 4;
    const int l16  = lane & 15;
    const int e0   = blockIdx.x * 16;

    v8f cw = {}, cb = {};
    const int arow = e0 + l16;
    const int cW   = h * 32 + l16;              // head h cols of Ee (256 wide)
    for (int k = 0; k < IN_DIM; k += 4) {
        const int ka = k + half * 2;
        v2f a;
        a.x = ea[arow * IN_DIM + ka];
        a.y = ea[arow * IN_DIM + ka + 1];
        v2f b0, b1;
        b0.x = Ew[(ka    ) * 256 + cW];
        b0.y = Ew[(ka + 1) * 256 + cW];
        b1.x = Ew[(ka    ) * 256 + cW + 16];
        b1.y = Ew[(ka + 1) * 256 + cW + 16];
        cw = wmma_f32_k4(a, b0, cw);
        cb = wmma_f32_k4(a, b1, cb);
    }

    const float biasW = Eb[h * 32 + l16];
    const float biasB = Eb[h * 32 + 16 + l16];
    const int   d     = l16;                    // this lane's d index
    const float aw    = Aw[d * NH + h];         // Aw is (D, H, 1)

    float sc[8];
#pragma unroll
    for (int r = 0; r < 8; ++r) {
        const int e  = e0 + r + half * 8;
        const int sn = src[e];
        const int dn = dst[e];
        const float kv = K[sn * IN_DIM + h * HD + d];
        const float qv = Q[dn * IN_DIM + h * HD + d];
        const float ewv = cw[r] + biasW;
        const float ebv = cb[r] + biasB;
        const float s   = (kv + qv) * ewv;
        const float et  = ((s >= 0.0f) ? sqrtf(s) : -sqrtf(-s)) + ebv;
        wE[(long long)e * IN_DIM + h * HD + d] = et;
        sc[r] = et * aw;
    }

    // reduce each row's 16 d-lanes; lane l16==r finalizes row r
#pragma unroll
    for (int r = 0; r < 8; ++r) {
        float red = sc[r];
        red += __shfl_xor(red, 1, 16);
        red += __shfl_xor(red, 2, 16);
        red += __shfl_xor(red, 4, 16);
        red += __shfl_xor(red, 8, 16);
        if (l16 == r) {
            const int e  = e0 + r + half * 8;
            const int dn = dst[e];
            const float v  = fminf(fmaxf(red, -CLAMPV), CLAMPV);
            const float es = expf(v);
            score[e * NH + h] = es;
            atomicAdd(&ssum[dn * NH + h], es);
            if (h == 0) atomicAdd(&deg[dn], 1.0f);
        }
    }
}

// ---------------------------------------------------------------------------
// Scatter: wV[dst] += V[src]*w ; rowV[dst] += e_t*w  (w = softmax weight)
// ---------------------------------------------------------------------------
__global__ __launch_bounds__(256)
void scatter_kernel(const int* __restrict__ src, const int* __restrict__ dst,
                    const float* __restrict__ V,
                    const float* __restrict__ wE,
                    const float* __restrict__ score,
                    const float* __restrict__ ssum,
                    float* __restrict__ wV, float* __restrict__ rowV)
{
    const long long tid = (long long)blockIdx.x * blockDim.x + threadIdx.x;
    if (tid >= (long long)N_EDGES * IN_DIM) return;
    const int e = (int)(tid >> 7);
    const int c = (int)(tid & 127);
    const int h = c >> 4;
    const int dn = dst[e];
    const float w = score[e * NH + h] / (ssum[dn * NH + h] + 1e-16f);
    atomicAdd(&wV[dn * IN_DIM + c],   V[src[e] * IN_DIM + c] * w);
    atomicAdd(&rowV[dn * IN_DIM + c], wE[(long long)e * IN_DIM + c] * w);
}

// ---------------------------------------------------------------------------
// Final: wV += rowV @ VeRow[:,h,:], build cat(16x64) in LDS with per-node
// log-degree scales, h = cat @ proj_w + proj_b.  1 wave per (node-tile, head).
// ---------------------------------------------------------------------------
__global__ __launch_bounds__(256)
void final_kernel(const float* __restrict__ wVg, const float* __restrict__ rowV,
                  const float* __restrict__ deg,
                  const float* __restrict__ VeRow,
                  const float* __restrict__ Pw, const float* __restrict__ Pb,
                  float* __restrict__ hout)
{
    __shared__ float lds[NH][16 * 64];

    const int lane = threadIdx.x & 31;
    const int h    = threadIdx.x >> 5;
    const int half = lane >> 4;
    const int l16  = lane & 15;
    const int row0 = blockIdx.x * 16;

    // C init = wV tile
    v8f c;
#pragma unroll
    for (int r = 0; r < 8; ++r)
        c[r] = wVg[(row0 + r + half * 8) * IN_DIM + h * HD + l16];

    // c += rowV(16x16) @ VeRow[:,h,:](16x16)
    for (int k = 0; k < HD; k += 4) {
        const int ka = k + half * 2;
        v2f a, b;
        a.x = rowV[(row0 + l16) * IN_DIM + h * HD + ka];
        a.y = rowV[(row0 + l16) * IN_DIM + h * HD + ka + 1];
        b.x = VeRow[(ka    ) * (NH * HD) + h * HD + l16];
        b.y = VeRow[(ka + 1) * (NH * HD) + h * HD + l16];
        c = wmma_f32_k4(a, b, c);
    }

    // build cat tile: cols [0:16)=wV, [16:32)=wV*ld, [32:48)=wV/(1+ld), [48:64)=wV*(1+.5ld)
#pragma unroll
    for (int r = 0; r < 8; ++r) {
        const int rl = r + half * 8;
        const float ld = log1pf(deg[row0 + rl]);
        const float s1 = ld, s2 = 1.0f / (1.0f + ld), s3 = 1.0f + 0.5f * ld;
        lds[h][rl * 64 +      l16] = c[r];
        lds[h][rl * 64 + 16 + l16] = c[r] * s1;
        lds[h][rl * 64 + 32 + l16] = c[r] * s2;
        lds[h][rl * 64 + 48 + l16] = c[r] * s3;
    }
    __syncthreads();

    // h = cat(16x64) @ proj_w(64x16)
    v8f acc = {};
    for (int k = 0; k < 64; k += 4) {
        const int ka = k + half * 2;
        v2f a, b;
        a.x = lds[h][l16 * 64 + ka];
        a.y = lds[h][l16 * 64 + ka + 1];
        b.x = Pw[(ka    ) * HD + l16];
        b.y = Pw[(ka + 1) * HD + l16];
        acc = wmma_f32_k4(a, b, acc);
    }
    const float bias = Pb[l16];
#pragma unroll
    for (int r = 0; r < 8; ++r)
        hout[(row0 + r + half * 8) * IN_DIM + h * HD + l16] = acc[r] + bias;
}

// ---------------------------------------------------------------------------
extern "C" void kernel_launch(void* const* d_in, const int* in_sizes, int n_in,
                              void* d_out, int out_size, void* d_ws, size_t ws_size,
                              hipStream_t stream) {
    const float* x   = (const float*)d_in[0];
    const float* ea  = (const float*)d_in[1];
    const int*   ei  = (const int*)  d_in[2];
    const float* Qw  = (const float*)d_in[3];
    const float* Qb  = (const float*)d_in[4];
    const float* Kw  = (const float*)d_in[5];
    const float* Kb  = (const float*)d_in[6];
    const float* Vw  = (const float*)d_in[7];
    const float* Vb  = (const float*)d_in[8];
    const float* Ew  = (const float*)d_in[9];
    const float* Eb  = (const float*)d_in[10];
    const float* Aw  = (const float*)d_in[11];
    const float* VeR = (const float*)d_in[12];
    const float* Pw  = (const float*)d_in[13];
    const float* Pb  = (const float*)d_in[14];

    float* out  = (float*)d_out;
    float* hout = out;                                   // (N, H, D) flat
    float* wE   = out + (size_t)N_NODES * IN_DIM;        // (E, H*D) flat

    const size_t NODE_F = (size_t)N_NODES * IN_DIM;      // 6.4M
    float* ws    = (float*)d_ws;
    float* Qo    = ws;
    float* Ko    = Qo + NODE_F;
    float* Vo    = Ko + NODE_F;
    float* score = Vo + NODE_F;                          // E*8
    float* ssum  = score + (size_t)N_EDGES * NH;         // N*8
    float* deg   = ssum  + (size_t)N_NODES * NH;         // N
    float* wV    = deg   + (size_t)N_NODES;              // N*128
    float* rowV  = wV    + NODE_F;                       // N*128

    const int* src = ei;
    const int* dst = ei + N_EDGES;

    // zero ssum + deg + wV + rowV (contiguous region starting at ssum)
    const long long nz = (long long)N_NODES * NH + N_NODES + 2LL * NODE_F;
    zero_kernel<<<(unsigned)((nz + 255) / 256), 256, 0, stream>>>(ssum, nz);

    qkv_gemm<<<N_NODES / 16, 256, 0, stream>>>(x, Qw, Qb, Kw, Kb, Vw, Vb, Qo, Ko, Vo);

    edge_kernel<<<N_EDGES / 16, 256, 0, stream>>>(ea, src, dst, Ew, Eb, Aw,
                                                  Qo, Ko, wE, score, ssum, deg);

    const long long nsc = (long long)N_EDGES * IN_DIM;
    scatter_kernel<<<(unsigned)((nsc + 255) / 256), 256, 0, stream>>>(
        src, dst, Vo, wE, score, ssum, wV, rowV);

    final_kernel<<<N_NODES / 16, 256, 0, stream>>>(wV, rowV, deg, VeR, Pw, Pb, hout);
}